// GCN_48747878810191
// MI455X (gfx1250) — compile-verified
//
#include <hip/hip_runtime.h>
#include <hip/hip_bf16.h>

// ---------------------------------------------------------------------------
// GCN forward for MI455X (gfx1250, wave32).
//  - 6 GEMMs (100000x128x128) via V_WMMA_F32_16X16X4_F32 (fp32 matches ref).
//  - A panel + transposed W staged in LDS with pitch 132 (conflict-free b64
//    fragment loads; B fragment = one aligned ds_load_b64, no reg shuffles).
//  - Edge norm (dinv[src]*w*dinv[dst]) precomputed once, reused by 4 passes.
//  - Scatter-add via fp32 atomics (working set ~160MB fits the 192MB L2).
// ---------------------------------------------------------------------------

typedef __attribute__((ext_vector_type(2))) float v2f;
typedef __attribute__((ext_vector_type(8))) float v8f;

#define DIM 128
#define PITCH 132   // 132 mod 64 == 4 -> 32 lanes of a b64 frag load tile all 64 banks
#define BN_EPS 1e-5f

// ---------------- GEMM: out[MxD] = A[MxD] @ W[DxD] (+bias) (relu?) ----------
__global__ void gemm_wmma_f32(const float* __restrict__ A,
                              const float* __restrict__ W,
                              const float* __restrict__ bias,
                              float* __restrict__ out,
                              int M, int applyRelu) {
  extern __shared__ float lds[];
  float* Wt = lds;               // [n:128][k:PITCH]  (W transposed)
  float* Al = lds + DIM * PITCH; // [r:128][k:PITCH]  (A panel, 128 rows/block)

  const int tid = threadIdx.x;   // 256 threads = 8 waves
  const int rowBase = blockIdx.x * 128;

  // stage W transposed: coalesced float4 global reads, scatter into Wt[n][k]
  for (int i = tid * 4; i < DIM * DIM; i += 1024) {
    const int k = i >> 7;        // W row  (K index)
    const int n = i & (DIM - 1); // W col  (N index)
    float4 w = *(const float4*)(W + i);
    Wt[(n + 0) * PITCH + k] = w.x;
    Wt[(n + 1) * PITCH + k] = w.y;
    Wt[(n + 2) * PITCH + k] = w.z;
    Wt[(n + 3) * PITCH + k] = w.w;
  }
  // stage A panel: coalesced float4 reads, guarded for the partial last block
  for (int i = tid * 4; i < DIM * DIM; i += 1024) {
    const int r = i >> 7;
    const int c = i & (DIM - 1);
    const int row = rowBase + r;
    float4 a = (row < M) ? *(const float4*)(A + (size_t)row * DIM + c)
                         : make_float4(0.f, 0.f, 0.f, 0.f);
    *(float4*)(Al + r * PITCH + c) = a;   // (r*132+c)*4 is 16B aligned (c%4==0)
  }
  __syncthreads();

  const int wave = tid >> 5;                   // 0..7: 16-row stripe per wave
  const int lane = tid & 31;
  const int m0 = rowBase + wave * 16;
  if (m0 >= M) return;                         // uniform per-wave, EXEC all-1s

  const int hsel = lane >> 4;                  // 0: K+{0,1}, 1: K+{2,3}
  const int l16  = lane & 15;

  v8f acc[8] = {};                             // 8 N-tiles of 16 cols each

  const float* Arow = Al + (wave * 16 + l16) * PITCH;

#pragma unroll 4
  for (int kk = 0; kk < 32; ++kk) {
    const int k = kk * 4 + hsel * 2;
    v2f a = *(const v2f*)(Arow + k);           // one conflict-free ds_load_b64
#pragma unroll
    for (int nt = 0; nt < 8; ++nt) {
      v2f b = *(const v2f*)(Wt + (nt * 16 + l16) * PITCH + k); // ds_load_b64
      acc[nt] = __builtin_amdgcn_wmma_f32_16x16x4_f32(
          /*neg_a=*/false, a, /*neg_b=*/false, b,
          /*c_mod=*/(short)0, acc[nt], /*reuse_a=*/false, /*reuse_b=*/false);
    }
  }

  // epilogue: bias + optional relu, store 16x128 tile
#pragma unroll
  for (int nt = 0; nt < 8; ++nt) {
    const int col = nt * 16 + l16;
    const float bv = bias ? bias[col] : 0.f;
#pragma unroll
    for (int v = 0; v < 8; ++v) {
      const int row = m0 + v + hsel * 8;
      float val = acc[nt][v] + bv;
      if (applyRelu) val = fmaxf(val, 0.f);
      out[(size_t)row * DIM + col] = val;
    }
  }
}

// ---------------- degree / norm precompute ---------------------------------
__global__ void k_fill1(float* __restrict__ p, int n) {
  int i = blockIdx.x * blockDim.x + threadIdx.x;
  if (i < n) p[i] = 1.0f;                      // self-loop weight
}

__global__ void k_deg(float* __restrict__ deg, const int* __restrict__ dst,
                      const float* __restrict__ ew, int E) {
  int e = blockIdx.x * blockDim.x + threadIdx.x;
  if (e < E) atomicAdd(&deg[dst[e]], ew[e]);
}

__global__ void k_dinv(float* __restrict__ dinv, const float* __restrict__ deg, int n) {
  int i = blockIdx.x * blockDim.x + threadIdx.x;
  if (i < n) {
    float d = deg[i];
    dinv[i] = (d > 0.f) ? rsqrtf(fmaxf(d, 1e-12f)) : 0.f;
  }
}

__global__ void k_norm(float* __restrict__ nrm, const int* __restrict__ src,
                       const int* __restrict__ dst, const float* __restrict__ ew,
                       const float* __restrict__ dinv, int E) {
  int e = blockIdx.x * blockDim.x + threadIdx.x;
  if (e < E) nrm[e] = dinv[src[e]] * ew[e] * dinv[dst[e]];
}

// ---------------- aggregation ----------------------------------------------
__global__ void k_bias_init(float* __restrict__ out, const float* __restrict__ bias,
                            int total) {
  int i = blockIdx.x * blockDim.x + threadIdx.x;
  if (i < total) out[i] = bias[i & (DIM - 1)];
}

// one (edge, 4-col quad) per thread: gather float4, scale, 4 fp32 atomics
__global__ void k_scatter(const float* __restrict__ t, const float* __restrict__ nrm,
                          const int* __restrict__ src, const int* __restrict__ dst,
                          const float* __restrict__ dinv, float* __restrict__ out,
                          int E, int N) {
  int idx = blockIdx.x * blockDim.x + threadIdx.x;
  int e = idx >> 5;
  int q = idx & 31;
  if (e >= E + N) return;
  int s, d;
  float w;
  if (e < E) {
    s = src[e]; d = dst[e]; w = nrm[e];
  } else {                                     // self loop, weight 1
    s = d = e - E;
    float di = dinv[s];
    w = di * di;
  }
  float4 v = *(const float4*)(t + (size_t)s * DIM + q * 4);
  float* o = out + (size_t)d * DIM + q * 4;
  atomicAdd(o + 0, v.x * w);
  atomicAdd(o + 1, v.y * w);
  atomicAdd(o + 2, v.z * w);
  atomicAdd(o + 3, v.w * w);
}

// ---------------- batch norm -----------------------------------------------
__global__ void k_zero(float* __restrict__ p, int n) {
  int i = blockIdx.x * blockDim.x + threadIdx.x;
  if (i < n) p[i] = 0.f;
}

// 128 threads/block, thread d owns column d; block-strided rows, atomic merge
__global__ void k_bn_stats(const float* __restrict__ agg, float* __restrict__ sums,
                           int N) {
  int d = threadIdx.x;
  float s = 0.f, ss = 0.f;
  for (int r = blockIdx.x; r < N; r += gridDim.x) {
    float v = agg[(size_t)r * DIM + d];
    s += v;
    ss += v * v;
  }
  atomicAdd(&sums[d], s);
  atomicAdd(&sums[DIM + d], ss);
}

// h = relu((agg - mu) * rsqrt(var+eps) * g + b) + h   (skip connection)
__global__ void k_bn_apply(float* __restrict__ h, const float* __restrict__ agg,
                           const float* __restrict__ sums, const float* __restrict__ g,
                           const float* __restrict__ b, int total, float invN) {
  int i = blockIdx.x * blockDim.x + threadIdx.x;
  if (i >= total) return;
  int d = i & (DIM - 1);
  float mu = sums[d] * invN;
  float var = fmaxf(sums[DIM + d] * invN - mu * mu, 0.f);
  float v = (agg[i] - mu) * rsqrtf(var + BN_EPS) * g[d] + b[d];
  h[i] = fmaxf(v, 0.f) + h[i];
}

// ---------------------------------------------------------------------------
static inline int nblk(long long n, int b) { return (int)((n + b - 1) / b); }

extern "C" void kernel_launch(void* const* d_in, const int* in_sizes, int n_in,
                              void* d_out, int out_size, void* d_ws, size_t ws_size,
                              hipStream_t stream) {
  const float* x      = (const float*)d_in[0];
  const float* ew     = (const float*)d_in[1];
  const float* mlp_w1 = (const float*)d_in[2];
  const float* mlp_b1 = (const float*)d_in[3];
  const float* mlp_w2 = (const float*)d_in[4];
  const float* mlp_b2 = (const float*)d_in[5];
  const float* conv_w = (const float*)d_in[6];
  const float* conv_b = (const float*)d_in[7];
  const float* last_w = (const float*)d_in[8];
  const float* last_b = (const float*)d_in[9];
  const float* bn_g   = (const float*)d_in[10];
  const float* bn_b   = (const float*)d_in[11];
  const int*   eidx   = (const int*)d_in[12];

  const int N = in_sizes[0] / DIM;   // 100000
  const int E = in_sizes[1];         // 1600000
  const int L = 3;
  const int* src = eidx;
  const int* dst = eidx + E;

  float* ws = (float*)d_ws;
  const size_t ND = (size_t)N * DIM;
  float* t    = ws;                  // GEMM output h@W
  float* h    = ws + ND;             // running features / skip
  float* agg  = ws + 2 * ND;         // aggregation buffer
  float* nrm  = ws + 3 * ND;         // E floats
  float* deg  = nrm + E;             // N floats
  float* dinv = deg + N;             // N floats
  float* sums = dinv + N;            // 256 floats (sum, sumsq)

  const int mtiles = (N + 15) / 16;                    // 6250
  const dim3 gGemm((mtiles + 7) / 8);                  // 8 waves x 16 rows/block
  const size_t ldsBytes = 2u * DIM * PITCH * sizeof(float); // ~132KB dynamic LDS
  const int scatterThreads = (E + N) * 32;
  const float invN = 1.0f / (float)N;

  // pre-MLP: h = relu(x@W1+b1) @ W2 + b2
  gemm_wmma_f32<<<gGemm, 256, ldsBytes, stream>>>(x, mlp_w1, mlp_b1, t, N, 1);
  gemm_wmma_f32<<<gGemm, 256, ldsBytes, stream>>>(t, mlp_w2, mlp_b2, h, N, 0);

  // degree / dinv / per-edge norm (layer-invariant, computed once)
  k_fill1<<<nblk(N, 256), 256, 0, stream>>>(deg, N);
  k_deg<<<nblk(E, 256), 256, 0, stream>>>(deg, dst, ew, E);
  k_dinv<<<nblk(N, 256), 256, 0, stream>>>(dinv, deg, N);
  k_norm<<<nblk(E, 256), 256, 0, stream>>>(nrm, src, dst, ew, dinv, E);

  // 3 GCN blocks: conv -> BN -> relu -> +skip
  for (int i = 0; i < L; ++i) {
    gemm_wmma_f32<<<gGemm, 256, ldsBytes, stream>>>(
        h, conv_w + (size_t)i * DIM * DIM, nullptr, t, N, 0);
    k_bias_init<<<nblk((long long)ND, 256), 256, 0, stream>>>(agg, conv_b + i * DIM,
                                                              (int)ND);
    k_scatter<<<nblk(scatterThreads, 256), 256, 0, stream>>>(t, nrm, src, dst, dinv,
                                                             agg, E, N);
    k_zero<<<1, 256, 0, stream>>>(sums, 2 * DIM);
    k_bn_stats<<<1024, DIM, 0, stream>>>(agg, sums, N);
    k_bn_apply<<<nblk((long long)ND, 256), 256, 0, stream>>>(
        h, agg, sums, bn_g + i * DIM, bn_b + i * DIM, (int)ND, invN);
  }

  // last conv -> d_out
  float* out = (float*)d_out;
  gemm_wmma_f32<<<gGemm, 256, ldsBytes, stream>>>(h, last_w, nullptr, t, N, 0);
  k_bias_init<<<nblk((long long)ND, 256), 256, 0, stream>>>(out, last_b, (int)ND);
  k_scatter<<<nblk(scatterThreads, 256), 256, 0, stream>>>(t, nrm, src, dst, dinv,
                                                           out, E, N);
}